// FineRefinement_63685775065402
// MI455X (gfx1250) — compile-verified
//
#include <hip/hip_runtime.h>
#include <math.h>

typedef __attribute__((ext_vector_type(2))) float v2f;
typedef __attribute__((ext_vector_type(8))) float v8f;

#define NPTS 20000
#define CDIM 128
#define HF 640
#define WF 640
#define HWS (HF * WF)
#define HALFW 2
#define APAD 132   // padded LDS row stride for the A panel (bank-conflict free)

// workspace byte offsets (total ~10.8 MB)
#define OFF_G     0u          // 128*128 f32 = 65536 B
#define OFF_U     65536u      // 128 f32
#define OFF_BO0   66048u      // 20000 i32
#define OFF_BO1   146048u     // 20000 i32
#define OFF_VALID 226048u     // 20000 i32
#define OFF_H     524288u     // 20000*128 f32 = 10,240,000 B

// ---------------------------------------------------------------------------
// Kernel 1: per-keypoint coords, clamp, validity, pts0 output
// ---------------------------------------------------------------------------
__global__ void prep_kernel(const float* __restrict__ mk0,
                            const float* __restrict__ mk1,
                            int* __restrict__ bo0, int* __restrict__ bo1,
                            int* __restrict__ valid,
                            float* __restrict__ pts0) {
    int n = blockIdx.x * blockDim.x + threadIdx.x;
    if (n >= NPTS) return;
    float x0f = mk0[2 * n], y0f = mk0[2 * n + 1];
    float x1f = mk1[2 * n], y1f = mk1[2 * n + 1];
    // fine_ratio = stride/2 = 4 ; truncation toward zero, values >= 0
    int cx0 = (int)(x0f * 4.0f), cy0 = (int)(y0f * 4.0f);
    int cx1 = (int)(x1f * 4.0f), cy1 = (int)(y1f * 4.0f);
    int v = (cx0 >= HALFW) && (cx0 + HALFW < WF) && (cy0 >= HALFW) && (cy0 + HALFW < HF) &&
            (cx1 >= HALFW) && (cx1 + HALFW < WF) && (cy1 >= HALFW) && (cy1 + HALFW < HF);
    int cx0c = min(max(cx0, HALFW), WF - 1 - HALFW);
    int cy0c = min(max(cy0, HALFW), HF - 1 - HALFW);
    int cx1c = min(max(cx1, HALFW), WF - 1 - HALFW);
    int cy1c = min(max(cy1, HALFW), HF - 1 - HALFW);
    bo0[n] = cy0c * WF + cx0c;
    bo1[n] = cy1c * WF + cx1c;
    valid[n] = v;
    // pts0 = mk0 * stride + stride/2
    pts0[2 * n]     = x0f * 8.0f + 4.0f;
    pts0[2 * n + 1] = y0f * 8.0f + 4.0f;
}

// ---------------------------------------------------------------------------
// Kernel 2: G = W^T W  (128x128), WMMA f32 16x16x4.
// grid = 8 (M tiles), block = 256 = 8 waves, wave w -> N tile w.
// A[m=c1][k=o] = W[o][c1], B[k=o][n=c2] = W[o][c2]
// ---------------------------------------------------------------------------
__global__ void __launch_bounds__(256) gram_kernel(const float* __restrict__ Wm,
                                                   float* __restrict__ G) {
    int lane = threadIdx.x & 31;
    int wave = threadIdx.x >> 5;
    int lo = lane & 15;
    int hi = lane >> 4;           // 0: K={0,1}, 1: K={2,3} within a K-step
    int mbase = blockIdx.x * 16;
    int m = mbase + lo;           // c1
    int ncol = wave * 16 + lo;    // c2
    v8f acc = {};
#pragma unroll
    for (int k = 0; k < CDIM; k += 4) {
        int kk = k + 2 * hi;
        v2f a, b;
        a.x = Wm[kk * CDIM + m];
        a.y = Wm[(kk + 1) * CDIM + m];
        b.x = Wm[kk * CDIM + ncol];
        b.y = Wm[(kk + 1) * CDIM + ncol];
        acc = __builtin_amdgcn_wmma_f32_16x16x4_f32(false, a, false, b,
                                                    (short)0, acc, false, false);
    }
#pragma unroll
    for (int r = 0; r < 8; ++r) {
        G[(mbase + r + 8 * hi) * CDIM + ncol] = acc[r];
    }
}

// ---------------------------------------------------------------------------
// Kernel 3: u = W^T b (tiny)
// ---------------------------------------------------------------------------
__global__ void ubias_kernel(const float* __restrict__ Wm,
                             const float* __restrict__ b,
                             float* __restrict__ u) {
    int c = threadIdx.x;
    float s = 0.0f;
#pragma unroll 8
    for (int o = 0; o < CDIM; ++o) s += b[o] * Wm[o * CDIM + c];
    u[c] = s;
}

// ---------------------------------------------------------------------------
// Kernel 4: H[n][c2] = sum_c1 f0[c1, cy0[n], cx0[n]] * G[c1][c2] + u[c2]
// grid = 1250 (20000/16 M tiles), block = 256 = 8 waves (one per 16-col tile).
// G (64KB) and the gathered 16x128 A panel (8.25KB, padded) staged in LDS:
// the scattered f0 gather happens exactly once per block instead of 8x.
// ---------------------------------------------------------------------------
__global__ void __launch_bounds__(256) proj_gemm_kernel(const float* __restrict__ f0,
                                                        const float* __restrict__ Gm,
                                                        const float* __restrict__ u,
                                                        const int* __restrict__ bo0,
                                                        float* __restrict__ H) {
    __shared__ float sG[CDIM * CDIM];
    __shared__ float sA[16 * APAD];
    int mbase = blockIdx.x * 16;

    for (int i = threadIdx.x; i < CDIM * CDIM; i += 256) sG[i] = Gm[i];
    // cooperative gather of the A panel: 16 keypoints x 128 channels
    for (int i = threadIdx.x; i < 16 * CDIM; i += 256) {
        int m = i >> 7;       // keypoint row within tile
        int c = i & 127;      // channel
        sA[m * APAD + c] = f0[(size_t)c * HWS + bo0[mbase + m]];
    }
    __syncthreads();

    int lane = threadIdx.x & 31;
    int wave = threadIdx.x >> 5;
    int lo = lane & 15;
    int hi = lane >> 4;
    int ncol = wave * 16 + lo;
    const float* ap = &sA[lo * APAD];
    v8f acc = {};
#pragma unroll
    for (int k = 0; k < CDIM; k += 4) {
        int kk = k + 2 * hi;
        v2f a, b;
        a.x = ap[kk];                  // contiguous pair -> ds_load_b64
        a.y = ap[kk + 1];
        b.x = sG[kk * CDIM + ncol];
        b.y = sG[(kk + 1) * CDIM + ncol];
        acc = __builtin_amdgcn_wmma_f32_16x16x4_f32(false, a, false, b,
                                                    (short)0, acc, false, false);
    }
    float ub = u[ncol];
#pragma unroll
    for (int r = 0; r < 8; ++r) {
        H[(mbase + r + 8 * hi) * CDIM + ncol] = acc[r] + ub;
    }
}

// ---------------------------------------------------------------------------
// Kernel 5: corr_p = h[n] . f1_raw[:, cy1+dy, cx1+dx]; softmax; expected offset.
// One wave per keypoint; 8 waves/block x 2500 blocks = 20000 exactly.
// ---------------------------------------------------------------------------
__global__ void __launch_bounds__(256) corr_kernel(const float* __restrict__ f1,
                                                   const float* __restrict__ H,
                                                   const int* __restrict__ bo1,
                                                   const int* __restrict__ valid,
                                                   const float* __restrict__ mk1,
                                                   float* __restrict__ pts1) {
    __shared__ float sh[8][CDIM];
    int lane = threadIdx.x & 31;
    int wave = threadIdx.x >> 5;
    int n = blockIdx.x * 8 + wave;     // always < 20000

    // stage h[n] into LDS: 32 lanes x float4 = 128 floats
    const float4 hv = ((const float4*)(H + n * CDIM))[lane];
    sh[wave][lane * 4 + 0] = hv.x;
    sh[wave][lane * 4 + 1] = hv.y;
    sh[wave][lane * 4 + 2] = hv.z;
    sh[wave][lane * 4 + 3] = hv.w;
    __syncthreads();

    int p = lane;                       // patch position, 25 active
    float acc = -__builtin_inff();
    if (p < 25) {
        int dy = p / 5 - 2;
        int dx = p % 5 - 2;
        const float* fp = f1 + bo1[n] + dy * WF + dx;
        float s = 0.0f;
#pragma unroll 8
        for (int c = 0; c < CDIM; ++c) s += sh[wave][c] * fp[c * HWS];
        acc = s;
    }
    // wave32 max-reduce
    float mx = acc;
#pragma unroll
    for (int o = 16; o > 0; o >>= 1) mx = fmaxf(mx, __shfl_xor(mx, o, 32));

    float e  = (p < 25) ? __expf(acc - mx) : 0.0f;
    float gx = (p < 25) ? (float)(p % 5 - 2) : 0.0f;
    float gy = (p < 25) ? (float)(p / 5 - 2) : 0.0f;
    float se = e, sx = e * gx, sy = e * gy;
#pragma unroll
    for (int o = 16; o > 0; o >>= 1) {
        se += __shfl_xor(se, o, 32);
        sx += __shfl_xor(sx, o, 32);
        sy += __shfl_xor(sy, o, 32);
    }
    if (lane == 0) {
        float inv = 1.0f / se;
        int v = valid[n];
        float ox = v ? sx * inv * 2.0f : 0.0f;   // scale = stride / fine_ratio = 2
        float oy = v ? sy * inv * 2.0f : 0.0f;
        float x1 = mk1[2 * n], y1 = mk1[2 * n + 1];
        pts1[2 * n]     = x1 * 8.0f + 4.0f + ox;
        pts1[2 * n + 1] = y1 * 8.0f + 4.0f + oy;
    }
}

// ---------------------------------------------------------------------------
extern "C" void kernel_launch(void* const* d_in, const int* in_sizes, int n_in,
                              void* d_out, int out_size, void* d_ws, size_t ws_size,
                              hipStream_t stream) {
    const float* mk0 = (const float*)d_in[0];
    const float* mk1 = (const float*)d_in[1];
    const float* f0  = (const float*)d_in[2];
    const float* f1  = (const float*)d_in[3];
    const float* pw  = (const float*)d_in[4];
    const float* pb  = (const float*)d_in[5];
    // d_in[6] = stride (==8), hardcoded per setup_inputs

    char* ws = (char*)d_ws;
    float* G     = (float*)(ws + OFF_G);
    float* u     = (float*)(ws + OFF_U);
    int*   bo0   = (int*)(ws + OFF_BO0);
    int*   bo1   = (int*)(ws + OFF_BO1);
    int*   valid = (int*)(ws + OFF_VALID);
    float* Hbuf  = (float*)(ws + OFF_H);

    float* pts0 = (float*)d_out;
    float* pts1 = (float*)d_out + 2 * NPTS;

    prep_kernel<<<(NPTS + 255) / 256, 256, 0, stream>>>(mk0, mk1, bo0, bo1, valid, pts0);
    gram_kernel<<<8, 256, 0, stream>>>(pw, G);
    ubias_kernel<<<1, 128, 0, stream>>>(pw, pb, u);
    proj_gemm_kernel<<<NPTS / 16, 256, 0, stream>>>(f0, G, u, bo0, Hbuf);
    corr_kernel<<<NPTS / 8, 256, 0, stream>>>(f1, Hbuf, bo1, valid, mk1, pts1);
}